// VirtualNodeGraphPooler_19146964205613
// MI455X (gfx1250) — compile-verified
//
#include <hip/hip_runtime.h>
#include <hip/hip_bf16.h>
#include <stdint.h>

// ---------------------------------------------------------------------------
// VirtualNodeGraphPooler for MI455X (gfx1250).
//
// Pure concatenation / copy workload: ~564 MB read, ~551 MB written, 0 FLOPs.
// Bandwidth-bound at 23.3 TB/s HBM (~48us floor). Bulk copies use the CDNA5
// async global<->LDS streaming path (GLOBAL_LOAD_ASYNC_TO_LDS_B128 /
// GLOBAL_STORE_ASYNC_FROM_LDS_B128 + s_wait_asynccnt) with non-temporal
// cache policy, since every stream is touched exactly once and is larger
// than the 192 MB L2.
// ---------------------------------------------------------------------------

#define TPB 256   // 8 wave32 per block
#define UN  8     // 8 x float4 = 128B contiguous per lane per iteration

// gfx12-style CPol: TH in bits [2:0]; TH_NT == 1 (non-temporal).
#define VN_CPOL_NT 1

typedef int v4i __attribute__((ext_vector_type(4)));
#define VN_AS1 __attribute__((address_space(1)))
#define VN_AS3 __attribute__((address_space(3)))

#if defined(__gfx1250__) && \
    __has_builtin(__builtin_amdgcn_global_load_async_to_lds_b128) && \
    __has_builtin(__builtin_amdgcn_global_store_async_from_lds_b128)
#define HAVE_ASYNC_LDS 1
#else
#define HAVE_ASYNC_LDS 0
#endif

__device__ __forceinline__ void vn_wait_async0() {
#if defined(__gfx1250__)
#if __has_builtin(__builtin_amdgcn_s_wait_asynccnt)
    __builtin_amdgcn_s_wait_asynccnt(0);
#else
    asm volatile("s_wait_asynccnt 0" ::: "memory");
#endif
#endif
}

// Bulk streaming copy: global -> LDS -> global via the CDNA5 async path.
// Each lane owns UN private float4 LDS slots, so no cross-wave sharing and no
// barriers are needed; per-wave ordering is enforced with s_wait_asynccnt.
// (Partial ASYNCcnt waits are unsound across the load/store mix because the
// two classes complete out-of-order w.r.t. each other, so we amortize the
// two wait-0 fences over UN*16B per lane and rely on wave-level concurrency
// for latency hiding.)
__global__ void vn_copy_bulk(const float4* __restrict__ src,
                             float4* __restrict__ dst,
                             long long n4) {
#if HAVE_ASYNC_LDS
    __shared__ float4 stage[TPB * UN];   // 32 KB of the 320 KB/WGP LDS
    const long long t    = (long long)blockIdx.x * TPB + threadIdx.x;
    const long long step = (long long)gridDim.x * TPB * UN;
    for (long long base = t * UN; base < n4; base += step) {
        if (base + UN <= n4) {
            vn_wait_async0();  // prior stores have finished reading our slots
#pragma unroll
            for (int u = 0; u < UN; ++u) {
                __builtin_amdgcn_global_load_async_to_lds_b128(
                    (VN_AS1 v4i*)(src + base + u),
                    (VN_AS3 v4i*)(&stage[threadIdx.x * UN + u]),
                    0, VN_CPOL_NT);
            }
            vn_wait_async0();  // loads have landed in LDS
#pragma unroll
            for (int u = 0; u < UN; ++u) {
                __builtin_amdgcn_global_store_async_from_lds_b128(
                    (VN_AS1 v4i*)(dst + base + u),
                    (VN_AS3 v4i*)(&stage[threadIdx.x * UN + u]),
                    0, VN_CPOL_NT);
            }
        } else {
            vn_wait_async0();
            for (long long k = base; k < n4; ++k) dst[k] = src[k];
        }
    }
    vn_wait_async0();          // drain ASYNCcnt before wave exits
#else
    const long long t    = (long long)blockIdx.x * blockDim.x + threadIdx.x;
    const long long step = (long long)gridDim.x * blockDim.x;
    for (long long i = t; i < n4; i += step) dst[i] = src[i];
#endif
}

// Vectorized zero-fill (b128 stores).
__global__ void vn_zero4(float4* __restrict__ dst, long long n4) {
    const long long t    = (long long)blockIdx.x * blockDim.x + threadIdx.x;
    const long long step = (long long)gridDim.x * blockDim.x;
    const float4 z = make_float4(0.f, 0.f, 0.f, 0.f);
    for (long long i = t; i < n4; i += step) dst[i] = z;
}

// Scalar zero-fill fallback for non-/4 sizes.
__global__ void vn_zero1(float* __restrict__ dst, long long n) {
    const long long t    = (long long)blockIdx.x * blockDim.x + threadIdx.x;
    const long long step = (long long)gridDim.x * blockDim.x;
    for (long long i = t; i < n; i += step) dst[i] = 0.0f;
}

// new_edge_index, row-major [2, E+N]:
//   row0: edge_index[0, :E] then arange(N)
//   row1: edge_index[1, :E] then N + batch[node]
// row/col via compare (no 64-bit division); flat index math in int64, row/col in int32.
__global__ void vn_edge_index(const int* __restrict__ ei,
                              const int* __restrict__ batch,
                              float* __restrict__ out,
                              int E, int N) {
    const long long L    = (long long)E + N;
    const long long tot  = 2 * L;
    const long long t    = (long long)blockIdx.x * blockDim.x + threadIdx.x;
    const long long step = (long long)gridDim.x * blockDim.x;
    for (long long i = t; i < tot; i += step) {
        const int row = (i >= L) ? 1 : 0;
        const int col = (int)(i - (row ? L : 0));
        float v;
        if (col < E) {
            v = (float)ei[(long long)row * E + col];
        } else if (row == 0) {
            v = (float)(col - E);            // heads = arange(N)
        } else {
            v = (float)(N + batch[col - E]); // tails = N + batch
        }
        out[i] = v;
    }
}

// new_batch: batch then arange(B)
__global__ void vn_batch(const int* __restrict__ batch,
                         float* __restrict__ out, int N, int B) {
    const int t = blockIdx.x * blockDim.x + threadIdx.x;
    if (t < N + B) out[t] = (t < N) ? (float)batch[t] : (float)(t - N);
}

static unsigned vn_blocks(long long items, long long per_block) {
    long long b = (items + per_block - 1) / per_block;
    if (b < 1) b = 1;
    if (b > (1 << 20)) b = (1 << 20);
    return (unsigned)b;
}

static void vn_launch_zero(float* dst, long long n, hipStream_t stream) {
    if ((n & 3) == 0) {
        const long long n4 = n >> 2;
        vn_zero4<<<vn_blocks(n4, TPB), TPB, 0, stream>>>((float4*)dst, n4);
    } else {
        vn_zero1<<<vn_blocks(n, TPB), TPB, 0, stream>>>(dst, n);
    }
}

extern "C" void kernel_launch(void* const* d_in, const int* in_sizes, int n_in,
                              void* d_out, int out_size, void* d_ws, size_t ws_size,
                              hipStream_t stream) {
    (void)n_in; (void)out_size; (void)d_ws; (void)ws_size;

    const float* x     = (const float*)d_in[0];
    const int*   ei    = (const int*)d_in[1];
    const float* ea    = (const float*)d_in[2];
    const int*   batch = (const int*)d_in[3];
    float* out = (float*)d_out;

    const long long xN  = in_sizes[0];       // N*D
    const long long eiN = in_sizes[1];       // 2*E
    const long long eaN = in_sizes[2];       // E*De
    const long long N   = in_sizes[3];
    const long long E   = eiN / 2;
    const long long D   = xN / N;            // 256
    const long long De  = eaN / E;           // 32
    const long long B   = 128;               // N_GRAPHS (static in reference)

    float* out_x  = out;                              // (N+B)*D
    float* out_ei = out_x  + (N + B) * D;             // 2*(E+N)
    float* out_ea = out_ei + 2 * (E + N);             // (E+N)*De
    float* out_b  = out_ea + (E + N) * De;            // N+B

    // 1) new_x: copy x (N*D floats, /4 exact since D=256), zero virtual nodes.
    const long long x4 = xN / 4;
    vn_copy_bulk<<<vn_blocks(x4, (long long)TPB * UN), TPB, 0, stream>>>(
        (const float4*)x, (float4*)out_x, x4);
    vn_launch_zero(out_x + N * D, B * D, stream);

    // 2) new_edge_index: copy + generated virtual edges (int -> f32 exact).
    vn_edge_index<<<vn_blocks(2 * (E + N), TPB), TPB, 0, stream>>>(
        ei, batch, out_ei, (int)E, (int)N);

    // 3) new_edge_attribute: bulk copy (E*De floats, /4 exact since De=32),
    //    zero the N appended rows.
    const long long ea4 = eaN / 4;
    vn_copy_bulk<<<vn_blocks(ea4, (long long)TPB * UN), TPB, 0, stream>>>(
        (const float4*)ea, (float4*)out_ea, ea4);
    vn_launch_zero(out_ea + E * De, N * De, stream);

    // 4) new_batch: copy + arange(B).
    vn_batch<<<vn_blocks(N + B, TPB), TPB, 0, stream>>>(batch, out_b, (int)N, (int)B);
}